// AttentionLayer_41188736368660
// MI455X (gfx1250) — compile-verified
//
#include <hip/hip_runtime.h>
#include <math.h>

// CDNA5 / gfx1250, wave32. fp32 WMMA 16x16x4 for the small GEMMs,
// bandwidth-optimal streaming (non-temporal 128-bit) for the big passes.

typedef float v2f __attribute__((ext_vector_type(2)));
typedef float v4f __attribute__((ext_vector_type(4)));
typedef float v8f __attribute__((ext_vector_type(8)));

#define NB   16      // batches
#define NS   8192    // sequence
#define NC   256     // channels (= D_MODEL = K_CH = OUT)
#define SCALE 0.0625f // 1/sqrt(256)

// workspace layout (float offsets)
#define WS_QH    0                   // 16*256 : scale * (q @ Wk^T)
#define WS_QB    (WS_QH + NB*NC)     // 16     : scale * (q . bk)
#define WS_MX    (WS_QB + NB)        // 16     : rowwise max of scores
#define WS_Z     (WS_MX + NB)        // 16     : sum exp
#define WS_SC    (WS_Z + NB + 16)    // 16*8192: scores
#define WS_PART  (WS_SC + NB*NS)     // 16*16*256 partial weighted sums
#define WS_CTX   (WS_PART + NB*16*NC) // 16*256 final context

// ---------------------------------------------------------------------------
// Kernel 1: q = query@Wq + bq ; qh = scale*(q@Wk^T) ; qb = scale*(q.bk)
// One block, 16 waves; wave w owns output columns [16w, 16w+16).
// ---------------------------------------------------------------------------
__global__ __launch_bounds__(512) void k1_proj(const float* __restrict__ query,
                                               const float* __restrict__ Wq,
                                               const float* __restrict__ bq,
                                               const float* __restrict__ Wk,
                                               const float* __restrict__ bk,
                                               float* ws) {
  __shared__ float q[NB * NC];
  const int tid  = threadIdx.x;
  const int wv   = tid >> 5;
  const int lane = tid & 31;
  const int m    = lane & 15;           // A row
  const int kb   = (lane >> 4) * 2;     // A/B K sub-offset
  const int n    = lane & 15;           // B/D column within tile
  const int n0   = wv * 16;
  const int mrow = (lane < 16) ? 0 : 8; // D row base per lane half

  // ---- q = query @ Wq (+bq) via V_WMMA_F32_16X16X4_F32, K-loop over 256 ----
  v8f acc = {};
  for (int k0 = 0; k0 < NC; k0 += 4) {
    v2f a, b;
    a.x = query[m * NC + k0 + kb];
    a.y = query[m * NC + k0 + kb + 1];
    b.x = Wq[(k0 + kb) * NC + n0 + n];
    b.y = Wq[(k0 + kb + 1) * NC + n0 + n];
    acc = __builtin_amdgcn_wmma_f32_16x16x4_f32(false, a, false, b,
                                                (short)0, acc, false, false);
  }
  {
    const float bias = bq[n0 + n];
#pragma unroll
    for (int r = 0; r < 8; ++r)
      q[(mrow + r) * NC + n0 + n] = acc[r] + bias;
  }
  __syncthreads();

  // ---- qh = scale * q @ Wk^T  (B element (K=o,N=c) = Wk[c*256+o]) ----------
  v8f acc2 = {};
  for (int k0 = 0; k0 < NC; k0 += 4) {
    v2f a, b;
    a.x = q[m * NC + k0 + kb];
    a.y = q[m * NC + k0 + kb + 1];
    const int c = n0 + n;
    b.x = Wk[c * NC + k0 + kb];
    b.y = Wk[c * NC + k0 + kb + 1];
    acc2 = __builtin_amdgcn_wmma_f32_16x16x4_f32(false, a, false, b,
                                                 (short)0, acc2, false, false);
  }
#pragma unroll
  for (int r = 0; r < 8; ++r)
    ws[WS_QH + (mrow + r) * NC + n0 + n] = acc2[r] * SCALE;

  // ---- qb[b] = scale * q[b,:].bk ----
  if (tid < NB) {
    float s = 0.f;
    for (int o = 0; o < NC; ++o) s += q[tid * NC + o] * bk[o];
    ws[WS_QB + tid] = s * SCALE;
  }
}

// ---------------------------------------------------------------------------
// Kernel 2: scores[b,s] = key[b,s,:].qh[b] + qb[b].  One wave per row,
// 2x non-temporal 128-bit loads per lane (coalesced 1KB per wave),
// shfl_xor reduction. key is single-use -> TH=NT keeps L2 for hot state.
// ---------------------------------------------------------------------------
__global__ __launch_bounds__(256) void k2_scores(const float* __restrict__ key,
                                                 float* ws) {
  const int tid  = threadIdx.x;
  const int wv   = tid >> 5;
  const int lane = tid & 31;
  const int row  = blockIdx.x * 8 + wv;   // 0..131071
  const int b    = row >> 13;             // row / 8192
  const v4f* kp = (const v4f*)(key + (size_t)row * NC) + lane * 2;
  const v4f* qp = (const v4f*)(ws + WS_QH + b * NC) + lane * 2;
  const v4f k0 = __builtin_nontemporal_load(kp);
  const v4f k1 = __builtin_nontemporal_load(kp + 1);
  const v4f q0 = qp[0], q1 = qp[1];
  const v4f d = k0 * q0 + k1 * q1;
  float v = d.x + d.y + d.z + d.w;
#pragma unroll
  for (int off = 16; off > 0; off >>= 1) v += __shfl_xor(v, off, 32);
  if (lane == 0) ws[WS_SC + row] = v + ws[WS_QB + b];
}

// ---------------------------------------------------------------------------
// Kernel 3: per-batch softmax stats (max, sum exp). 16 blocks.
// ---------------------------------------------------------------------------
__global__ __launch_bounds__(256) void k3_stats(float* ws) {
  __shared__ float red[256];
  const int b = blockIdx.x, tid = threadIdx.x;
  const float* sc = ws + WS_SC + b * NS;
  float m = -3.0e38f;
  for (int i = tid; i < NS; i += 256) m = fmaxf(m, sc[i]);
  red[tid] = m; __syncthreads();
  for (int s = 128; s > 0; s >>= 1) {
    if (tid < s) red[tid] = fmaxf(red[tid], red[tid + s]);
    __syncthreads();
  }
  m = red[0]; __syncthreads();
  float sum = 0.f;
  for (int i = tid; i < NS; i += 256) sum += expf(sc[i] - m);
  red[tid] = sum; __syncthreads();
  for (int s = 128; s > 0; s >>= 1) {
    if (tid < s) red[tid] += red[tid + s];
    __syncthreads();
  }
  if (tid == 0) { ws[WS_MX + b] = m; ws[WS_Z + b] = red[0]; }
}

// ---------------------------------------------------------------------------
// Kernel 4: partial vctx[b,c] += softmax_w * value[b,s,c] over 512-row chunks.
// 256 blocks (b x 16 chunks). Deterministic fixed-order sums, no atomics.
// value is single-use -> non-temporal 128-bit loads.
// ---------------------------------------------------------------------------
__global__ __launch_bounds__(256) void k4_wsum(const float* __restrict__ value,
                                               float* ws) {
  __shared__ float p[512];
  __shared__ float accl[4 * 256];
  const int b   = blockIdx.x >> 4;
  const int ch  = blockIdx.x & 15;
  const int s0  = ch * 512;
  const int tid = threadIdx.x;
  const float mx   = ws[WS_MX + b];
  const float invZ = 1.0f / ws[WS_Z + b];
  for (int i = tid; i < 512; i += 256)
    p[i] = expf(ws[WS_SC + b * NS + s0 + i] - mx) * invZ;
  __syncthreads();

  const int g = tid >> 6;   // row-group 0..3
  const int t = tid & 63;   // 128-bit channel slot
  v4f acc = {};
  const v4f* vp = (const v4f*)(value + ((size_t)b * NS + s0) * NC) + t;
  for (int s = g; s < 512; s += 4) {
    const v4f v = __builtin_nontemporal_load(vp + s * 64); // row = 64 v4f
    acc += p[s] * v;
  }
  ((v4f*)accl)[g * 64 + t] = acc;
  __syncthreads();
  const float sum = accl[0 * 256 + tid] + accl[1 * 256 + tid]
                  + accl[2 * 256 + tid] + accl[3 * 256 + tid];
  ws[WS_PART + blockIdx.x * 256 + tid] = sum;
}

// ---------------------------------------------------------------------------
// Kernel 5: vctx = sum partials; ctx = vctx @ Wv + bv (WMMA f32 16x16x4).
// ---------------------------------------------------------------------------
__global__ __launch_bounds__(512) void k5_ctx(const float* __restrict__ Wv,
                                              const float* __restrict__ bv,
                                              float* ws) {
  __shared__ float vctx[NB * NC];
  const int tid = threadIdx.x;
  for (int idx = tid; idx < NB * NC; idx += 512) {
    const int b = idx >> 8, c = idx & 255;
    float s = 0.f;
#pragma unroll
    for (int ch = 0; ch < 16; ++ch) s += ws[WS_PART + (b * 16 + ch) * 256 + c];
    vctx[idx] = s;
  }
  __syncthreads();

  const int wv   = tid >> 5;
  const int lane = tid & 31;
  const int m    = lane & 15;
  const int kb   = (lane >> 4) * 2;
  const int n    = lane & 15;
  const int n0   = wv * 16;
  const int mrow = (lane < 16) ? 0 : 8;
  v8f acc = {};
  for (int k0 = 0; k0 < NC; k0 += 4) {
    v2f a, b;
    a.x = vctx[m * NC + k0 + kb];
    a.y = vctx[m * NC + k0 + kb + 1];
    b.x = Wv[(k0 + kb) * NC + n0 + n];
    b.y = Wv[(k0 + kb + 1) * NC + n0 + n];
    acc = __builtin_amdgcn_wmma_f32_16x16x4_f32(false, a, false, b,
                                                (short)0, acc, false, false);
  }
  const float bias = bv[n0 + n];
#pragma unroll
  for (int r = 0; r < 8; ++r)
    ws[WS_CTX + (mrow + r) * NC + n0 + n] = acc[r] + bias;
}

// ---------------------------------------------------------------------------
// Kernel 6: out[b,s,:] = ctx[b,:], non-temporal 128-bit stores (128 MiB,
// write-once stream -> do not dirty L2).
// ---------------------------------------------------------------------------
__global__ __launch_bounds__(256) void k6_bcast(const float* __restrict__ ws,
                                                float* __restrict__ out) {
  const size_t idx = (size_t)blockIdx.x * 256 + threadIdx.x; // v4f index
  const int    o4  = (int)(idx & 63);                        // 64 v4f / row
  const size_t row = idx >> 6;
  const int    b   = (int)(row >> 13);                       // row / 8192
  const v4f v = ((const v4f*)(ws + WS_CTX))[b * 64 + o4];
  __builtin_nontemporal_store(v, (v4f*)out + idx);
}

// ---------------------------------------------------------------------------
extern "C" void kernel_launch(void* const* d_in, const int* in_sizes, int n_in,
                              void* d_out, int out_size, void* d_ws, size_t ws_size,
                              hipStream_t stream) {
  const float* query = (const float*)d_in[0];
  const float* key   = (const float*)d_in[1];
  const float* value = (const float*)d_in[2];
  const float* Wq    = (const float*)d_in[3];
  const float* bq    = (const float*)d_in[4];
  const float* Wk    = (const float*)d_in[5];
  const float* bk    = (const float*)d_in[6];
  const float* Wv    = (const float*)d_in[7];
  const float* bv    = (const float*)d_in[8];
  float* ws  = (float*)d_ws;
  float* out = (float*)d_out;

  k1_proj  <<<1,            512, 0, stream>>>(query, Wq, bq, Wk, bk, ws);
  k2_scores<<<NB * NS / 8,  256, 0, stream>>>(key, ws);
  k3_stats <<<NB,           256, 0, stream>>>(ws);
  k4_wsum  <<<NB * 16,      256, 0, stream>>>(value, ws);
  k5_ctx   <<<1,            512, 0, stream>>>(Wv, bv, ws);
  k6_bcast <<<(NB * NS * NC / 4) / 256, 256, 0, stream>>>(ws, out);
}